// ProposalLayer_40467181863113
// MI455X (gfx1250) — compile-verified
//
#include <hip/hip_runtime.h>
#include <hip/hip_bf16.h>

// ---------------- problem constants ----------------
#define HH 128
#define WW 192
#define AA 9
#define HW (HH * WW)              // 24576
#define NTOT (HW * AA)            // 221184
#define PRE 6000
#define POST 300
#define WORDS 188                 // ceil(6000/32)
#define NMS_T 0.7f
#define SORT_N 8192

typedef float v8f __attribute__((ext_vector_type(8)));
typedef float v2f __attribute__((ext_vector_type(2)));

// 9 base anchors (base=16, ratios {0.5,1,2}, scales {8,16,32}) — precomputed
// exactly as _generate_anchors() produces them.
__constant__ float c_anchors[9][4] = {
    {-84.f,  -40.f,   99.f,  55.f},
    {-176.f, -88.f,  191.f, 103.f},
    {-360.f, -184.f, 375.f, 199.f},
    {-56.f,  -56.f,   71.f,  71.f},
    {-120.f, -120.f, 135.f, 135.f},
    {-248.f, -248.f, 263.f, 263.f},
    {-36.f,  -80.f,   51.f,  95.f},
    {-80.f,  -168.f,  95.f, 183.f},
    {-168.f, -344.f, 183.f, 359.f}};

static __device__ __forceinline__ int imin(int a, int b) { return a < b ? a : b; }

// monotonic float->uint key (ascending uint == ascending float)
static __device__ __forceinline__ unsigned orderKey(float f) {
    unsigned u = __float_as_uint(f);
    return (u & 0x80000000u) ? ~u : (u | 0x80000000u);
}
#define NEG_INF_KEY 0x007FFFFFu   // orderKey(-inf)

static __device__ __forceinline__ unsigned wave_ballot(bool p) {
#if __has_builtin(__builtin_amdgcn_ballot_w32)
    return (unsigned)__builtin_amdgcn_ballot_w32(p);
#else
    return (unsigned)__ballot(p);
#endif
}

// ---------------- K0: init ----------------
__global__ void k0_init(unsigned* hist, unsigned* meta,
                        unsigned long long* cand, int* sel, float* out) {
    int t = blockIdx.x * blockDim.x + threadIdx.x;  // 65536 threads
    hist[t] = 0u;
    if (t < 64) meta[t] = 0u;
    if (t < SORT_N) cand[t] = ~0ull;                // sorts to the end (ascending)
    if (t < 512) sel[t] = -1;
    if (t < POST * 5) out[t] = 0.0f;
}

// ---------------- K1: proposal decode + masked score key ----------------
__global__ void k1_proposals(const float* __restrict__ scores,
                             const float* __restrict__ deltas,
                             const float* __restrict__ im_info,
                             float4* __restrict__ props,
                             unsigned* __restrict__ key32) {
    int i = blockIdx.x * blockDim.x + threadIdx.x;
    if (i >= NTOT) return;
    int a = i % AA;
    int pix = i / AA;
    int w = pix % WW;
    int h = pix / WW;

    float sc = scores[(AA + a) * HW + h * WW + w];
    float dx = deltas[(4 * a + 0) * HW + h * WW + w];
    float dy = deltas[(4 * a + 1) * HW + h * WW + w];
    float dw = deltas[(4 * a + 2) * HW + h * WW + w];
    float dh = deltas[(4 * a + 3) * HW + h * WW + w];
    dw = fminf(fmaxf(dw, -10.0f), 10.0f);
    dh = fminf(fmaxf(dh, -10.0f), 10.0f);

    float sx = (float)(w * 16);
    float sy = (float)(h * 16);
    float ax1 = c_anchors[a][0] + sx, ay1 = c_anchors[a][1] + sy;
    float ax2 = c_anchors[a][2] + sx, ay2 = c_anchors[a][3] + sy;

    float bw = ax2 - ax1 + 1.0f;
    float bh = ay2 - ay1 + 1.0f;
    float cx = ax1 + 0.5f * bw;
    float cy = ay1 + 0.5f * bh;

    float pcx = dx * bw + cx;
    float pcy = dy * bh + cy;
    float pw = expf(dw) * bw;
    float ph = expf(dh) * bh;

    float x1 = pcx - 0.5f * pw, y1 = pcy - 0.5f * ph;
    float x2 = pcx + 0.5f * pw, y2 = pcy + 0.5f * ph;

    float imH = im_info[0], imW = im_info[1], imS = im_info[2];
    x1 = fminf(fmaxf(x1, 0.0f), imW - 1.0f);
    x2 = fminf(fmaxf(x2, 0.0f), imW - 1.0f);
    y1 = fminf(fmaxf(y1, 0.0f), imH - 1.0f);
    y2 = fminf(fmaxf(y2, 0.0f), imH - 1.0f);

    float msz = 16.0f * imS;
    bool valid = (x2 - x1 + 1.0f >= msz) && (y2 - y1 + 1.0f >= msz);
    float masked = valid ? sc : -__builtin_inff();

    props[i] = make_float4(x1, y1, x2, y2);
    key32[i] = orderKey(masked);
}

// ---------------- K2: 16-bit-bucket histogram ----------------
__global__ void k2_hist(const unsigned* __restrict__ key32, unsigned* hist) {
    int i = blockIdx.x * blockDim.x + threadIdx.x;
    if (i >= NTOT) return;
    atomicAdd(&hist[key32[i] >> 16], 1u);
}

// ---------------- K3: pick threshold bucket B (top-PRE cut) ----------------
__global__ void k3_bucket(const unsigned* __restrict__ hist, unsigned* meta) {
    __shared__ unsigned chunkSum[256];
    int t = threadIdx.x;
    unsigned s = 0;
    for (int k = 0; k < 256; ++k) s += hist[t * 256 + k];
    chunkSum[t] = s;
    __syncthreads();
    if (t == 0) {
        unsigned acc = 0;
        int B = 0;
        bool found = false;
        for (int c = 255; c >= 0 && !found; --c) {
            if (acc + chunkSum[c] >= PRE) {
                for (int b = c * 256 + 255; b >= c * 256; --b) {
                    acc += hist[b];
                    if (acc >= PRE) { B = b; found = true; break; }
                }
            } else {
                acc += chunkSum[c];
            }
        }
        meta[1] = (unsigned)B;
    }
}

// ---------------- K4: compact candidates (key bucket >= B) ----------------
__global__ void k4_compact(const unsigned* __restrict__ key32,
                           unsigned* meta, unsigned long long* cand) {
    int i = blockIdx.x * blockDim.x + threadIdx.x;
    if (i >= NTOT) return;
    unsigned B = meta[1];
    unsigned k = key32[i];
    if ((k >> 16) >= B) {
        unsigned pos = atomicAdd(&meta[0], 1u);
        if (pos < SORT_N) {
            // ascending sort => descending score, ties broken by ascending index
            cand[pos] = (((unsigned long long)(0xFFFFFFFFu - k)) << 32) |
                        (unsigned long long)(unsigned)i;
        }
    }
}

// ---------------- K5: single-WGP LDS bitonic sort (8192 x u64) + gather ----------------
__global__ void __launch_bounds__(1024)
k5_sort_gather(const unsigned long long* __restrict__ cand,
               const float4* __restrict__ props,
               float4* __restrict__ topBoxes,
               unsigned* __restrict__ tvalid) {
    __shared__ unsigned long long skey[SORT_N];   // 64 KB resident in LDS
    int t = threadIdx.x;
    for (int i = t; i < SORT_N; i += 1024) skey[i] = cand[i];
    __syncthreads();
    for (int k = 2; k <= SORT_N; k <<= 1) {
        for (int j = k >> 1; j > 0; j >>= 1) {
            for (int idx = t; idx < SORT_N; idx += 1024) {
                int ixj = idx ^ j;
                if (ixj > idx) {
                    bool up = ((idx & k) == 0);
                    unsigned long long x = skey[idx], y = skey[ixj];
                    if ((x > y) == up) { skey[idx] = y; skey[ixj] = x; }
                }
            }
            __syncthreads();
        }
    }
    for (int r = t; r < PRE; r += 1024) {
        unsigned long long key = skey[r];
        unsigned id = (unsigned)(key & 0xFFFFFFFFull);
        unsigned k32 = 0xFFFFFFFFu - (unsigned)(key >> 32);
        if (id >= NTOT) id = 0;
        topBoxes[r] = props[id];
        tvalid[r] = (k32 > NEG_INF_KEY) ? 1u : 0u;
    }
}

// ---------------- K6: suppression mask via WMMA 16x16x4 f32 + ballot ----------------
// Block = 1 wave (32 threads). Tile = 16 rows x 32 cols of the 6000x6000 matrix.
// area_i + area_j for a 16x16 tile is a rank-2 outer product -> v_wmma_f32_16x16x4_f32.
static __device__ __forceinline__ float boxArea(float4 b) {
    return (b.z - b.x) * (b.w - b.y);
}

__global__ void __launch_bounds__(32)
k6_mask(const float4* __restrict__ topBoxes, unsigned* __restrict__ maskBuf) {
    __shared__ float4 sRow[16];
    __shared__ float4 sCol[32];
    int lane = threadIdx.x;
    int rowBase = blockIdx.y * 16;
    int colBase = blockIdx.x * 32;

    if (lane < 16) sRow[lane] = topBoxes[imin(rowBase + lane, PRE - 1)];
    sCol[lane] = topBoxes[imin(colBase + lane, PRE - 1)];
    __syncthreads();

    int l16 = lane & 15;
    bool lo = lane < 16;
    float areaR = boxArea(sRow[l16]);
    float areaC1 = boxArea(sCol[l16]);
    float areaC2 = boxArea(sCol[16 + l16]);

    // A (16x4): lanes 0-15 hold K=0,1 ; lanes 16-31 hold K=2,3 (zeroed)
    v2f Av;  Av.x  = lo ? areaR : 0.0f;  Av.y  = lo ? 1.0f : 0.0f;
    // B (4x16): lanes 0-15 hold K=0,1 per column N=lane
    v2f B1v; B1v.x = lo ? 1.0f : 0.0f;   B1v.y = lo ? areaC1 : 0.0f;
    v2f B2v; B2v.x = lo ? 1.0f : 0.0f;   B2v.y = lo ? areaC2 : 0.0f;

    v8f cz = {0.0f, 0.0f, 0.0f, 0.0f, 0.0f, 0.0f, 0.0f, 0.0f};
    // D[M][N] = area_row[M] + area_col[N]
    v8f d1 = __builtin_amdgcn_wmma_f32_16x16x4_f32(false, Av, false, B1v,
                                                   (short)0, cz, false, false);
    v8f d2 = __builtin_amdgcn_wmma_f32_16x16x4_f32(false, Av, false, B2v,
                                                   (short)0, cz, false, false);

    bool colOk1 = (colBase + l16) < PRE;
    bool colOk2 = (colBase + 16 + l16) < PRE;

    unsigned b1[8], b2[8];
#pragma unroll
    for (int r = 0; r < 8; ++r) {
        float4 rb = sRow[r + (lo ? 0 : 8)];   // M = r (lanes 0-15) or r+8 (lanes 16-31)
        float4 cb1 = sCol[l16];
        float4 cb2 = sCol[16 + l16];

        float iw1 = fminf(rb.z, cb1.z) - fmaxf(rb.x, cb1.x);
        float ih1 = fminf(rb.w, cb1.w) - fmaxf(rb.y, cb1.y);
        float in1 = fmaxf(iw1, 0.0f) * fmaxf(ih1, 0.0f);
        float un1 = d1[r] - in1;                       // a_i + a_j - inter
        bool p1 = (in1 > NMS_T * un1) && colOk1;
        b1[r] = wave_ballot(p1);

        float iw2 = fminf(rb.z, cb2.z) - fmaxf(rb.x, cb2.x);
        float ih2 = fminf(rb.w, cb2.w) - fmaxf(rb.y, cb2.y);
        float in2 = fmaxf(iw2, 0.0f) * fmaxf(ih2, 0.0f);
        float un2 = d2[r] - in2;
        bool p2 = (in2 > NMS_T * un2) && colOk2;
        b2[r] = wave_ballot(p2);
    }

    // ballot bit L: lanes 0-15 -> row M=r cols 0-15, lanes 16-31 -> row M=r+8
    if (lane < 16) {
        int row = rowBase + lane;
        if (row < PRE) {
            unsigned wrd;
            if (lane < 8) {
                wrd = (b1[lane] & 0xFFFFu) | ((b2[lane] & 0xFFFFu) << 16);
            } else {
                int r = lane - 8;
                wrd = (b1[r] >> 16) | (b2[r] & 0xFFFF0000u);
            }
            maskBuf[(size_t)row * WORDS + blockIdx.x] = wrd;
        }
    }
}

// ---------------- K7: sequential greedy NMS scan (single block) ----------------
__global__ void k7_scan(const unsigned* __restrict__ tvalid,
                        const unsigned* __restrict__ maskBuf,
                        int* __restrict__ sel) {
    __shared__ unsigned removed[WORDS];
    __shared__ int decide;
    __shared__ int kcount;
    int t = threadIdx.x;
    if (t < WORDS) removed[t] = 0u;
    if (t == 0) kcount = 0;
    __syncthreads();
    for (int i = 0; i < PRE; ++i) {
        if (t == 0) {
            unsigned w = removed[i >> 5];
            int dec = (tvalid[i] != 0u) && !((w >> (i & 31)) & 1u);
            decide = dec;
            if (dec) {
                if (kcount < POST) sel[kcount] = i;
                kcount++;
            }
        }
        __syncthreads();
        if (decide && t < WORDS) removed[t] |= maskBuf[(size_t)i * WORDS + t];
        __syncthreads();
    }
}

// ---------------- K8: write output (300 x 5), rows pre-zeroed ----------------
__global__ void k8_output(const int* __restrict__ sel,
                          const float4* __restrict__ topBoxes,
                          float* __restrict__ out) {
    int r = threadIdx.x;
    if (r >= POST) return;
    int i = sel[r];
    if (i >= 0) {
        float4 b = topBoxes[i];
        out[r * 5 + 0] = 0.0f;
        out[r * 5 + 1] = b.x;
        out[r * 5 + 2] = b.y;
        out[r * 5 + 3] = b.z;
        out[r * 5 + 4] = b.w;
    }
}

// ---------------- host launch ----------------
extern "C" void kernel_launch(void* const* d_in, const int* in_sizes, int n_in,
                              void* d_out, int out_size, void* d_ws, size_t ws_size,
                              hipStream_t stream) {
    const float* scores  = (const float*)d_in[0];
    const float* deltas  = (const float*)d_in[1];
    const float* im_info = (const float*)d_in[2];
    float* out = (float*)d_out;

    char* ws = (char*)d_ws;
    size_t off = 0;
    auto carve = [&](size_t bytes) -> char* {
        char* p = ws + off;
        off = (off + bytes + 255) & ~(size_t)255;
        return p;
    };
    float4*             props    = (float4*)carve((size_t)NTOT * 16);
    unsigned*           key32    = (unsigned*)carve((size_t)NTOT * 4);
    unsigned*           hist     = (unsigned*)carve(65536 * 4);
    unsigned*           meta     = (unsigned*)carve(64 * 4);
    unsigned long long* cand     = (unsigned long long*)carve(SORT_N * 8);
    float4*             topBoxes = (float4*)carve((size_t)PRE * 16);
    unsigned*           tvalid   = (unsigned*)carve((size_t)PRE * 4);
    int*                sel      = (int*)carve(512 * 4);
    unsigned*           maskBuf  = (unsigned*)carve((size_t)PRE * WORDS * 4);

    k0_init<<<256, 256, 0, stream>>>(hist, meta, cand, sel, out);
    k1_proposals<<<NTOT / 256, 256, 0, stream>>>(scores, deltas, im_info, props, key32);
    k2_hist<<<NTOT / 256, 256, 0, stream>>>(key32, hist);
    k3_bucket<<<1, 256, 0, stream>>>(hist, meta);
    k4_compact<<<NTOT / 256, 256, 0, stream>>>(key32, meta, cand);
    k5_sort_gather<<<1, 1024, 0, stream>>>(cand, props, topBoxes, tvalid);
    k6_mask<<<dim3(WORDS, PRE / 16), 32, 0, stream>>>(topBoxes, maskBuf);
    k7_scan<<<1, 256, 0, stream>>>(tvalid, maskBuf, sel);
    k8_output<<<1, 512, 0, stream>>>(sel, topBoxes, out);
}